// SGC_20117626814729
// MI455X (gfx1250) — compile-verified
//
#include <hip/hip_runtime.h>
#include <math.h>

#define N_NODES  100000
#define DIM_IN   128
#define DIM_HID  64
#define DIM_OUT  40
#define N_EDGES  1600000
#define M_TILES  (N_NODES / 16)   // 6250 exactly

typedef __attribute__((ext_vector_type(2))) float v2f;
typedef __attribute__((ext_vector_type(8))) float v8f;

// ---------------------------------------------------------------- degree pass
__global__ void init_deg_kernel(float* __restrict__ deg) {
    int i = blockIdx.x * blockDim.x + threadIdx.x;
    if (i < N_NODES) deg[i] = 1.0f;   // self-loop
}

__global__ void deg_accum_kernel(const long long* __restrict__ dst,
                                 float* __restrict__ deg) {
    int e = blockIdx.x * blockDim.x + threadIdx.x;
    if (e < N_EDGES) atomicAdd(&deg[(int)dst[e]], 1.0f);
}

__global__ void dinv_kernel(float* __restrict__ deg) {
    int i = blockIdx.x * blockDim.x + threadIdx.x;
    if (i < N_NODES) deg[i] = rsqrtf(deg[i]);
}

// ---------------------------------------------------------------- GEMM1: h1 = x @ W1   [100000x128]x[128x64]
// W1 staged in LDS (32 KB of the WGP's 320 KB); B-fragments come from ds_load,
// keeping the VMEM pipe free for the streaming A (x) traffic that hits HBM.
__global__ void __launch_bounds__(256)
gemm1_kernel(const float* __restrict__ X, const float* __restrict__ W,
             float* __restrict__ H) {
    __shared__ float sW[DIM_IN * DIM_HID];                 // 8192 floats = 32 KB
    for (int i = threadIdx.x; i < DIM_IN * DIM_HID; i += 256)
        sW[i] = W[i];
    __syncthreads();

    const int wave = (int)(blockIdx.x * 8 + (threadIdx.x >> 5));
    const int lane = threadIdx.x & 31;
    if (wave >= M_TILES) return;           // wave-uniform exit; EXEC stays all-ones
    const int half = lane >> 4;            // 0 or 1
    const int ln   = lane & 15;
    const int m0   = wave * 16;
    const float* __restrict__ xrow = X + (size_t)(m0 + ln) * DIM_IN;

    v8f acc0 = {}, acc1 = {}, acc2 = {}, acc3 = {};
#pragma unroll 4
    for (int k0 = 0; k0 < DIM_IN; k0 += 4) {
        const int ka = k0 + 2 * half;
        v2f a;  a.x = xrow[ka];  a.y = xrow[ka + 1];
        const float* w0 = sW + ka * DIM_HID;
        const float* w1 = w0 + DIM_HID;
        v2f b0; b0.x = w0[ln     ]; b0.y = w1[ln     ];
        v2f b1; b1.x = w0[ln + 16]; b1.y = w1[ln + 16];
        v2f b2; b2.x = w0[ln + 32]; b2.y = w1[ln + 32];
        v2f b3; b3.x = w0[ln + 48]; b3.y = w1[ln + 48];
        acc0 = __builtin_amdgcn_wmma_f32_16x16x4_f32(false, a, false, b0, (short)0, acc0, false, false);
        acc1 = __builtin_amdgcn_wmma_f32_16x16x4_f32(false, a, false, b1, (short)0, acc1, false, false);
        acc2 = __builtin_amdgcn_wmma_f32_16x16x4_f32(false, a, false, b2, (short)0, acc2, false, false);
        acc3 = __builtin_amdgcn_wmma_f32_16x16x4_f32(false, a, false, b3, (short)0, acc3, false, false);
    }
    float* __restrict__ o = H + (size_t)m0 * DIM_HID;
#pragma unroll
    for (int i = 0; i < 8; ++i) {
        const int r = i + 8 * half;        // C/D layout: VGPR i -> row i (lanes 0-15) / i+8 (lanes 16-31)
        o[(size_t)r * DIM_HID + ln     ] = acc0[i];
        o[(size_t)r * DIM_HID + ln + 16] = acc1[i];
        o[(size_t)r * DIM_HID + ln + 32] = acc2[i];
        o[(size_t)r * DIM_HID + ln + 48] = acc3[i];
    }
}

// ---------------------------------------------------------------- layer1 aggregation
__global__ void self1_kernel(const float* __restrict__ h1,
                             const float* __restrict__ dinv,
                             const float* __restrict__ bias,
                             float* __restrict__ agg) {
    int idx = blockIdx.x * blockDim.x + threadIdx.x;
    if (idx < N_NODES * DIM_HID) {
        int node = idx >> 6, f = idx & 63;
        float di = dinv[node];
        agg[idx] = di * di * h1[idx] + bias[f];
    }
}

__global__ void scatter1_kernel(const long long* __restrict__ src,
                                const long long* __restrict__ dst,
                                const float* __restrict__ dinv,
                                const float* __restrict__ h1,
                                float* __restrict__ agg) {
    int wave = (int)((blockIdx.x * blockDim.x + threadIdx.x) >> 5);
    int lane = threadIdx.x & 31;
    if (wave >= N_EDGES) return;
    int s = (int)src[wave], d = (int)dst[wave];
    float w = dinv[s] * dinv[d];
    const float2* __restrict__ hs = (const float2*)(h1 + (size_t)s * DIM_HID);
    float*        __restrict__ ad = agg + (size_t)d * DIM_HID;
    float2 v = hs[lane];                       // one coalesced b64 gather per lane
    atomicAdd(&ad[2 * lane    ], v.x * w);
    atomicAdd(&ad[2 * lane + 1], v.y * w);
}

// ---------------------------------------------------------------- GEMM2: h2 = agg1 @ W2   [100000x64]x[64x40]
__global__ void __launch_bounds__(256)
gemm2_kernel(const float* __restrict__ A, const float* __restrict__ W,
             float* __restrict__ H) {
    __shared__ float sW[DIM_HID * DIM_OUT];                // 2560 floats = 10 KB
    for (int i = threadIdx.x; i < DIM_HID * DIM_OUT; i += 256)
        sW[i] = W[i];
    __syncthreads();

    const int wave = (int)(blockIdx.x * 8 + (threadIdx.x >> 5));
    const int lane = threadIdx.x & 31;
    if (wave >= M_TILES) return;
    const int half = lane >> 4;
    const int ln   = lane & 15;
    const int m0   = wave * 16;
    const float* __restrict__ arow = A + (size_t)(m0 + ln) * DIM_HID;

    v8f acc0 = {}, acc1 = {}, acc2 = {};
#pragma unroll 4
    for (int k0 = 0; k0 < DIM_HID; k0 += 4) {
        const int ka = k0 + 2 * half;
        v2f a;  a.x = arow[ka];  a.y = arow[ka + 1];
        const float* w0 = sW + ka * DIM_OUT;
        const float* w1 = w0 + DIM_OUT;
        v2f b0; b0.x = w0[ln     ]; b0.y = w1[ln     ];
        v2f b1; b1.x = w0[ln + 16]; b1.y = w1[ln + 16];
        v2f b2;                                       // cols 32..47, only 32..39 valid
        b2.x = (ln + 32 < DIM_OUT) ? w0[ln + 32] : 0.0f;
        b2.y = (ln + 32 < DIM_OUT) ? w1[ln + 32] : 0.0f;
        acc0 = __builtin_amdgcn_wmma_f32_16x16x4_f32(false, a, false, b0, (short)0, acc0, false, false);
        acc1 = __builtin_amdgcn_wmma_f32_16x16x4_f32(false, a, false, b1, (short)0, acc1, false, false);
        acc2 = __builtin_amdgcn_wmma_f32_16x16x4_f32(false, a, false, b2, (short)0, acc2, false, false);
    }
    float* __restrict__ o = H + (size_t)m0 * DIM_OUT;
#pragma unroll
    for (int i = 0; i < 8; ++i) {
        const int r = i + 8 * half;
        o[(size_t)r * DIM_OUT + ln     ] = acc0[i];
        o[(size_t)r * DIM_OUT + ln + 16] = acc1[i];
        if (ln + 32 < DIM_OUT)
            o[(size_t)r * DIM_OUT + ln + 32] = acc2[i];
    }
}

// ---------------------------------------------------------------- layer2 aggregation
__global__ void self2_kernel(const float* __restrict__ h2,
                             const float* __restrict__ dinv,
                             const float* __restrict__ bias,
                             float* __restrict__ agg) {
    int idx = blockIdx.x * blockDim.x + threadIdx.x;
    if (idx < N_NODES * DIM_OUT) {
        int node = idx / DIM_OUT, f = idx % DIM_OUT;
        float di = dinv[node];
        agg[idx] = di * di * h2[idx] + bias[f];
    }
}

__global__ void scatter2_kernel(const long long* __restrict__ src,
                                const long long* __restrict__ dst,
                                const float* __restrict__ dinv,
                                const float* __restrict__ h2,
                                float* __restrict__ agg) {
    int wave = (int)((blockIdx.x * blockDim.x + threadIdx.x) >> 5);
    int lane = threadIdx.x & 31;
    if (wave >= N_EDGES) return;
    int s = (int)src[wave], d = (int)dst[wave];
    float w = dinv[s] * dinv[d];
    const float2* __restrict__ hs = (const float2*)(h2 + (size_t)s * DIM_OUT);
    float*        __restrict__ ad = agg + (size_t)d * DIM_OUT;
    if (lane < DIM_OUT / 2) {                  // 20 float2 per edge
        float2 v = hs[lane];
        atomicAdd(&ad[2 * lane    ], v.x * w);
        atomicAdd(&ad[2 * lane + 1], v.y * w);
    }
}

// ---------------------------------------------------------------- log_softmax (per node)
__global__ void logsoftmax_kernel(const float* __restrict__ agg,
                                  float* __restrict__ out) {
    int i = blockIdx.x * blockDim.x + threadIdx.x;
    if (i >= N_NODES) return;
    const float2* __restrict__ a = (const float2*)(agg + (size_t)i * DIM_OUT);
    float2 buf[DIM_OUT / 2];
    float mx = -INFINITY;
#pragma unroll
    for (int f = 0; f < DIM_OUT / 2; ++f) {
        buf[f] = a[f];
        mx = fmaxf(mx, fmaxf(buf[f].x, buf[f].y));
    }
    float s = 0.0f;
#pragma unroll
    for (int f = 0; f < DIM_OUT / 2; ++f)
        s += __expf(buf[f].x - mx) + __expf(buf[f].y - mx);
    float lse = __logf(s) + mx;
    float2* __restrict__ o = (float2*)(out + (size_t)i * DIM_OUT);
#pragma unroll
    for (int f = 0; f < DIM_OUT / 2; ++f) {
        float2 r;  r.x = buf[f].x - lse;  r.y = buf[f].y - lse;
        o[f] = r;
    }
}

// ---------------------------------------------------------------- launcher
extern "C" void kernel_launch(void* const* d_in, const int* in_sizes, int n_in,
                              void* d_out, int out_size, void* d_ws, size_t ws_size,
                              hipStream_t stream) {
    const float*     x   = (const float*)d_in[0];
    const long long* ei  = (const long long*)d_in[1];
    const float*     W1  = (const float*)d_in[2];
    const float*     b1  = (const float*)d_in[3];
    const float*     W2  = (const float*)d_in[4];
    const float*     b2  = (const float*)d_in[5];
    float*           out = (float*)d_out;

    const long long* src = ei;
    const long long* dst = ei + N_EDGES;

    // workspace layout (floats): dinv | h1 | agg1 | h2 | agg2  (~83.6 MB)
    float* ws    = (float*)d_ws;
    float* dinv  = ws;                       // 100000 (padded to 100096)
    float* h1    = ws + 100096;              // 6,400,000
    float* agg1  = h1 + (size_t)N_NODES * DIM_HID;
    float* h2    = agg1 + (size_t)N_NODES * DIM_HID;
    float* agg2  = h2 + (size_t)N_NODES * DIM_OUT;

    const int T = 256;
    // degree / normalization
    init_deg_kernel <<<(N_NODES + T - 1) / T, T, 0, stream>>>(dinv);
    deg_accum_kernel<<<(N_EDGES + T - 1) / T, T, 0, stream>>>(dst, dinv);
    dinv_kernel     <<<(N_NODES + T - 1) / T, T, 0, stream>>>(dinv);

    // layer 1
    const int gemm_blocks = (M_TILES + 7) / 8;   // 8 waves/block, 1 M-tile/wave
    gemm1_kernel   <<<gemm_blocks, T, 0, stream>>>(x, W1, h1);
    self1_kernel   <<<(N_NODES * DIM_HID + T - 1) / T, T, 0, stream>>>(h1, dinv, b1, agg1);
    scatter1_kernel<<<(N_EDGES * 32) / T, T, 0, stream>>>(src, dst, dinv, h1, agg1);

    // layer 2
    gemm2_kernel   <<<gemm_blocks, T, 0, stream>>>(agg1, W2, h2);
    self2_kernel   <<<(N_NODES * DIM_OUT + T - 1) / T, T, 0, stream>>>(h2, dinv, b2, agg2);
    scatter2_kernel<<<(N_EDGES * 32) / T, T, 0, stream>>>(src, dst, dinv, h2, agg2);

    // output
    logsoftmax_kernel<<<(N_NODES + T - 1) / T, T, 0, stream>>>(agg2, out);
}